// EntityAlignmentModule_33741263077482
// MI455X (gfx1250) — compile-verified
//
#include <hip/hip_runtime.h>

// ---------------------------------------------------------------------------
// EntityAlignmentModule on gfx1250 (MI455X), fp32 end-to-end.
// GEMMs use native fp32 WMMA: v_wmma_f32_16x16x4_f32.
// ---------------------------------------------------------------------------

typedef float v2f __attribute__((ext_vector_type(2)));
typedef float v8f __attribute__((ext_vector_type(8)));

#define DM 512          // model dim D
#define B_SZ 32
#define E_SZ 64
#define R_SZ 36
#define SCORES_PER_B (E_SZ * R_SZ)   // 2304

__device__ __forceinline__ float wave_sum(float v) {
    #pragma unroll
    for (int m = 16; m >= 1; m >>= 1) v += __shfl_xor(v, m, 32);
    return v;
}

// ---------------------------------------------------------------------------
// Fused GEMM main loop (shared by both GEMM kernels).
// Workgroup: 16 rows (M) x 512 cols (N). 8 waves; wave w owns N in
// [w*64, w*64+64) as four 16x16 WMMA tiles. A chunk (16x32) staged in LDS.
//
// fp32 WMMA fragment layout (ISA 7.12.2):
//   A 16x4:  lane L -> M = L&15 ; VGPR v -> K = v + 2*(L>>4)
//   B 4x16:  lane L -> N = L&15 ; VGPR v -> K = v + 2*(L>>4)
//   C 16x16: lane L -> N = L&15 ; VGPR i -> M = i + 8*(L>>4)
// ---------------------------------------------------------------------------
#define GEMM_MAIN_LOOP(A, W, K, ldsA, acc0, acc1, acc2, acc3)                   \
    for (int kk = 0; kk < (K); kk += 32) {                                      \
        __syncthreads();                                                        \
        {   /* stage A[m0..m0+15][kk..kk+31] : 512 floats, 2 per thread */      \
            int i = tid * 2;                                                    \
            int r = i >> 5, c = i & 31;                                         \
            const float2 av = *(const float2*)((A) + (size_t)(m0 + r) * (K) + kk + c); \
            (ldsA)[i] = av.x; (ldsA)[i + 1] = av.y;                             \
        }                                                                       \
        __syncthreads();                                                        \
        _Pragma("unroll")                                                       \
        for (int ks = 0; ks < 8; ++ks) {                                        \
            v2f afrag;                                                          \
            afrag.x = (ldsA)[arow * 32 + ks * 4 + khalf + 0];                   \
            afrag.y = (ldsA)[arow * 32 + ks * 4 + khalf + 1];                   \
            const float* wp = (W) + (size_t)(kk + ks * 4 + khalf) * DM;         \
            const int nlan = n0 + (lane & 15);                                  \
            v2f bfrag;                                                          \
            bfrag.x = wp[nlan +  0]; bfrag.y = wp[DM + nlan +  0];              \
            acc0 = __builtin_amdgcn_wmma_f32_16x16x4_f32(false, afrag, false, bfrag, (short)0, acc0, false, false); \
            bfrag.x = wp[nlan + 16]; bfrag.y = wp[DM + nlan + 16];              \
            acc1 = __builtin_amdgcn_wmma_f32_16x16x4_f32(false, afrag, false, bfrag, (short)0, acc1, false, false); \
            bfrag.x = wp[nlan + 32]; bfrag.y = wp[DM + nlan + 32];              \
            acc2 = __builtin_amdgcn_wmma_f32_16x16x4_f32(false, afrag, false, bfrag, (short)0, acc2, false, false); \
            bfrag.x = wp[nlan + 48]; bfrag.y = wp[DM + nlan + 48];              \
            acc3 = __builtin_amdgcn_wmma_f32_16x16x4_f32(false, afrag, false, bfrag, (short)0, acc3, false, false); \
        }                                                                       \
    }

// out = LayerNorm(relu(A @ W + bias)) * gamma + beta, row-wise over DM.
__global__ void __launch_bounds__(256)
gemm_relu_ln_kernel(const float* __restrict__ A, const float* __restrict__ W,
                    const float* __restrict__ bias, const float* __restrict__ gamma,
                    const float* __restrict__ beta, float* __restrict__ out, int K)
{
    __shared__ float ldsA[16 * 32];
    __shared__ float ldsC[16 * DM];
    __shared__ float mu_s[16], rs_s[16];

    const int tid  = threadIdx.x;
    const int wave = tid >> 5;
    const int lane = tid & 31;
    const int m0   = blockIdx.x * 16;
    const int n0   = wave * 64;
    const int arow  = lane & 15;
    const int khalf = (lane >> 4) * 2;

    v8f acc0 = {}; v8f acc1 = {}; v8f acc2 = {}; v8f acc3 = {};
    GEMM_MAIN_LOOP(A, W, K, ldsA, acc0, acc1, acc2, acc3)

    // Epilogue 1: relu(acc + bias) -> LDS tile (16 x 512)
    const int chi = (lane >> 4) * 8;           // C row offset per lane half
    #pragma unroll
    for (int t = 0; t < 4; ++t) {
        const v8f acc = (t == 0) ? acc0 : (t == 1) ? acc1 : (t == 2) ? acc2 : acc3;
        const int col = n0 + t * 16 + (lane & 15);
        const float bv = bias[col];
        #pragma unroll
        for (int i = 0; i < 8; ++i) {
            const int row = i + chi;
            ldsC[row * DM + col] = fmaxf(acc[i] + bv, 0.0f);
        }
    }
    __syncthreads();

    // Epilogue 2: per-row mean/var (wave handles 2 rows, lane-parallel)
    #pragma unroll
    for (int rr = 0; rr < 2; ++rr) {
        const int row = wave * 2 + rr;
        float s = 0.f, s2 = 0.f;
        #pragma unroll
        for (int j = 0; j < 16; ++j) {
            const float x = ldsC[row * DM + lane + j * 32];
            s += x; s2 = fmaf(x, x, s2);
        }
        s = wave_sum(s); s2 = wave_sum(s2);
        if (lane == 0) {
            const float mu  = s * (1.0f / DM);
            const float var = s2 * (1.0f / DM) - mu * mu;
            mu_s[row] = mu;
            rs_s[row] = rsqrtf(var + 1e-5f);
        }
    }
    __syncthreads();

    // Epilogue 3: normalized, affine write-out
    #pragma unroll 4
    for (int j = 0; j < 32; ++j) {
        const int idx = tid + j * 256;
        const int row = idx >> 9, col = idx & (DM - 1);
        const float x = ldsC[idx];
        out[(size_t)(m0 + row) * DM + col] =
            (x - mu_s[row]) * rs_s[row] * gamma[col] + beta[col];
    }
}

// out = A @ W (+ bias if non-null); plain fp32 WMMA GEMM.
__global__ void __launch_bounds__(256)
gemm_bias_kernel(const float* __restrict__ A, const float* __restrict__ W,
                 const float* __restrict__ bias, float* __restrict__ out, int K)
{
    __shared__ float ldsA[16 * 32];

    const int tid  = threadIdx.x;
    const int wave = tid >> 5;
    const int lane = tid & 31;
    const int m0   = blockIdx.x * 16;
    const int n0   = wave * 64;
    const int arow  = lane & 15;
    const int khalf = (lane >> 4) * 2;

    v8f acc0 = {}; v8f acc1 = {}; v8f acc2 = {}; v8f acc3 = {};
    GEMM_MAIN_LOOP(A, W, K, ldsA, acc0, acc1, acc2, acc3)

    const int chi = (lane >> 4) * 8;
    #pragma unroll
    for (int t = 0; t < 4; ++t) {
        const v8f acc = (t == 0) ? acc0 : (t == 1) ? acc1 : (t == 2) ? acc2 : acc3;
        const int col = n0 + t * 16 + (lane & 15);
        const float bv = bias ? bias[col] : 0.0f;
        #pragma unroll
        for (int i = 0; i < 8; ++i) {
            const int row = i + chi;
            out[(size_t)(m0 + row) * DM + col] = acc[i] + bv;
        }
    }
}

// scores[b,e,r] = sigmoid( sum_d relu(ht[b,e,d] + hi[b,r,d]) * W2[d] + b2 )
// One block per (b,e); ht row and W2 cached in LDS; wave per r.
__global__ void __launch_bounds__(256)
score_kernel(const float* __restrict__ ht, const float* __restrict__ hi,
             const float* __restrict__ W2, const float* __restrict__ b2,
             float* __restrict__ out_scores)
{
    __shared__ float lds_ht[DM];
    __shared__ float lds_w2[DM];

    const int tid  = threadIdx.x;
    const int wave = tid >> 5;
    const int lane = tid & 31;
    const int be   = blockIdx.x;          // b*64 + e
    const int b    = be >> 6;

    {
        const float2 v = *(const float2*)(ht + (size_t)be * DM + tid * 2);
        lds_ht[tid * 2] = v.x; lds_ht[tid * 2 + 1] = v.y;
        const float2 w = *(const float2*)(W2 + tid * 2);
        lds_w2[tid * 2] = w.x; lds_w2[tid * 2 + 1] = w.y;
    }
    __syncthreads();

    const float bb = b2[0];
    for (int r = wave; r < R_SZ; r += 8) {
        const float* hp = hi + ((size_t)b * R_SZ + r) * DM;
        float p = 0.f;
        #pragma unroll
        for (int j = 0; j < 16; ++j) {
            const int d = lane + j * 32;
            const float v = fmaxf(lds_ht[d] + hp[d], 0.0f);
            p = fmaf(v, lds_w2[d], p);
        }
        p = wave_sum(p);
        if (lane == 0) {
            const float s = p + bb;
            out_scores[(size_t)be * R_SZ + r] = 1.0f / (1.0f + expf(-s));
        }
    }
}

// Per batch: softmax over E*R scores, fold over E, weighted sum of image_aligned.
__global__ void __launch_bounds__(256)
softmax_agg_kernel(const float* __restrict__ scores, const float* __restrict__ img,
                   float* __restrict__ out_agg)
{
    __shared__ float lds_s[SCORES_PER_B];
    __shared__ float red[8];
    __shared__ float coeff[R_SZ];
    __shared__ float m_sh, z_sh;

    const int tid  = threadIdx.x;
    const int wave = tid >> 5;
    const int lane = tid & 31;
    const int b    = blockIdx.x;
    const float* sp = scores + (size_t)b * SCORES_PER_B;

    float lmax = -1e30f;
    for (int i = tid; i < SCORES_PER_B; i += 256) {
        const float v = sp[i];
        lds_s[i] = v;
        lmax = fmaxf(lmax, v);
    }
    __syncthreads();
    #pragma unroll
    for (int m = 16; m >= 1; m >>= 1) lmax = fmaxf(lmax, __shfl_xor(lmax, m, 32));
    if (lane == 0) red[wave] = lmax;
    __syncthreads();
    if (tid == 0) {
        float m = red[0];
        #pragma unroll
        for (int i = 1; i < 8; ++i) m = fmaxf(m, red[i]);
        m_sh = m;
    }
    __syncthreads();
    const float m = m_sh;

    float ls = 0.f;
    for (int i = tid; i < SCORES_PER_B; i += 256) ls += expf(lds_s[i] - m);
    ls = wave_sum(ls);
    if (lane == 0) red[wave] = ls;
    __syncthreads();
    if (tid == 0) {
        float z = 0.f;
        #pragma unroll
        for (int i = 0; i < 8; ++i) z += red[i];
        z_sh = z;
    }
    __syncthreads();
    const float z = z_sh;

    // coeff[r] = (sum_e exp(score[e,r]-m)) / (Z * E)
    if (tid < R_SZ) {
        float s = 0.f;
        for (int e = 0; e < E_SZ; ++e) s += expf(lds_s[e * R_SZ + tid] - m);
        coeff[tid] = s / (z * (float)E_SZ);
    }
    __syncthreads();

    for (int d = tid; d < DM; d += 256) {
        float a = 0.f;
        #pragma unroll
        for (int r = 0; r < R_SZ; ++r)
            a = fmaf(coeff[r], img[((size_t)b * R_SZ + r) * DM + d], a);
        out_agg[(size_t)b * DM + d] = a;
    }
}

extern "C" void kernel_launch(void* const* d_in, const int* in_sizes, int n_in,
                              void* d_out, int out_size, void* d_ws, size_t ws_size,
                              hipStream_t stream) {
    const float* text_feats = (const float*)d_in[0];   // (32,64,768)
    const float* image_feats = (const float*)d_in[1];  // (32,36,2048)
    const float* W_text = (const float*)d_in[2];       // (768,512)
    const float* b_text = (const float*)d_in[3];
    const float* g_text = (const float*)d_in[4];
    const float* beta_text = (const float*)d_in[5];
    const float* W_img = (const float*)d_in[6];        // (2048,512)
    const float* b_img = (const float*)d_in[7];
    const float* g_img = (const float*)d_in[8];
    const float* beta_img = (const float*)d_in[9];
    const float* W1 = (const float*)d_in[10];          // (1024,512)
    const float* b1 = (const float*)d_in[11];
    const float* W2 = (const float*)d_in[12];          // (512,1)
    const float* b2 = (const float*)d_in[13];
    float* out = (float*)d_out;

    // Workspace layout (floats): ta | ia | ht | hi  (~12.5 MB total)
    float* ta = (float*)d_ws;                    // 2048*512
    float* ia = ta + 2048 * 512;                 // 1152*512
    float* ht = ia + 1152 * 512;                 // 2048*512
    float* hi = ht + 2048 * 512;                 // 1152*512

    // 1) text_aligned = LN(relu(text @ W_text + b_text))       M=2048, K=768
    gemm_relu_ln_kernel<<<128, 256, 0, stream>>>(text_feats, W_text, b_text,
                                                 g_text, beta_text, ta, 768);
    // 2) image_aligned = LN(relu(image @ W_img + b_img))       M=1152, K=2048
    gemm_relu_ln_kernel<<<72, 256, 0, stream>>>(image_feats, W_img, b_img,
                                                g_img, beta_img, ia, 2048);
    // 3) h_t = text_aligned @ W1[:512]                         M=2048, K=512
    gemm_bias_kernel<<<128, 256, 0, stream>>>(ta, W1, nullptr, ht, 512);
    // 4) h_i = image_aligned @ W1[512:] + b1                   M=1152, K=512
    gemm_bias_kernel<<<72, 256, 0, stream>>>(ia, W1 + 512 * 512, b1, hi, 512);
    // 5) alignment_scores (fused relu-add-dot + sigmoid) -> out[0 : 73728]
    score_kernel<<<B_SZ * E_SZ, 256, 0, stream>>>(ht, hi, W2, b2, out);
    // 6) softmax + aggregation -> out[73728 : 73728+16384]
    softmax_agg_kernel<<<B_SZ, 256, 0, stream>>>(out, ia, out + B_SZ * E_SZ * R_SZ);
}